// GCNN_RNN_22222160790111
// MI455X (gfx1250) — compile-verified
//
#include <hip/hip_runtime.h>
#include <hip/hip_bf16.h>

typedef __attribute__((ext_vector_type(2))) float v2f;
typedef __attribute__((ext_vector_type(8))) float v8f;

// Problem constants (match reference)
static constexpr int   NN    = 10000;            // nodes
static constexpr int   TT    = 8;                // input horizon
static constexpr int   PP    = 2;                // prediction horizon (zeros)
static constexpr int   FIN   = 64;
static constexpr int   FOUT  = 64;
static constexpr int   HH    = 128;
static constexpr int   BB    = 2;
static constexpr int   EE    = 160000;
static constexpr int   GG    = BB * TT;          // 16 graphs
static constexpr int   BN    = BB * NN;          // 20000 sequences
static constexpr int   HID   = HH * 4;           // 512
static constexpr size_t GNF  = (size_t)GG * NN * FOUT;   // 10.24M floats

__device__ __forceinline__ v8f wmma4(v2f a, v2f b, v8f c) {
  // D = A(16x4,f32) * B(4x16,f32) + C(16x16,f32)
  return __builtin_amdgcn_wmma_f32_16x16x4_f32(false, a, false, b, (short)0, c,
                                               false, false);
}

__device__ __forceinline__ void atomAddF(float* p, float v) {
  __hip_atomic_fetch_add(p, v, __ATOMIC_RELAXED, __HIP_MEMORY_SCOPE_AGENT);
}

// ---------------- degree / norm ----------------
__global__ void k_deg_init(float* deg) {
  int i = blockIdx.x * 256 + threadIdx.x;
  if (i < NN) deg[i] = 0.f;
}

__global__ void k_deg_scatter(const int* __restrict__ ei,
                              const float* __restrict__ ew,
                              float* __restrict__ deg) {
  int e = blockIdx.x * 256 + threadIdx.x;
  if (e < EE)                atomAddF(&deg[ei[EE + e]], ew[e]);   // col
  else if (e < EE + NN)      atomAddF(&deg[e - EE], 1.0f);        // self loop
}

__global__ void k_dinv(const float* __restrict__ deg, float* __restrict__ dinv) {
  int i = blockIdx.x * 256 + threadIdx.x;
  if (i < NN) {
    float d = deg[i];
    dinv[i] = d > 0.f ? rsqrtf(d) : 0.f;
  }
}

// ---------------- GCN linear: xl = x @ gcn_w.T (160000x64 @ 64x64) ----------------
__global__ __launch_bounds__(256) void k_gcn_gemm(const float* __restrict__ X,
                                                  const float* __restrict__ W,
                                                  float* __restrict__ Y) {
  __shared__ float wl[FOUT * FIN];                 // 16 KB
  for (int i = threadIdx.x; i < FOUT * FIN; i += 256) wl[i] = W[i];
  __syncthreads();
  const int wave = threadIdx.x >> 5, lane = threadIdx.x & 31;
  const int mr = lane & 15, half = lane >> 4, kk = half * 2;
  const int row0 = blockIdx.x * 32 + (wave & 1) * 16;
  const int n0 = (wave >> 1) * 16;
  const float* ap = X + (size_t)(row0 + mr) * FIN + kk;
  const float* bp = wl + (n0 + mr) * FIN + kk;     // B[k][n] = W[n][k]
  v8f c = {};
#pragma unroll
  for (int k = 0; k < FIN; k += 4) {
    v2f a = {ap[k], ap[k + 1]};
    v2f b = {bp[k], bp[k + 1]};
    c = wmma4(a, b, c);
  }
#pragma unroll
  for (int j = 0; j < 8; ++j)
    Y[(size_t)(row0 + j + 8 * half) * FOUT + n0 + mr] = c[j];
}

// ---------------- init gc with bias ----------------
__global__ void k_agg_init(const float* __restrict__ b, float* __restrict__ gc) {
  size_t i = (size_t)blockIdx.x * 256 + threadIdx.x;
  if (i < GNF) gc[i] = b[i & 63];
}

// ---------------- edge scatter-add over all 16 graphs ----------------
__global__ void k_scatter(const int* __restrict__ ei, const float* __restrict__ ew,
                          const float* __restrict__ dinv,
                          const float* __restrict__ xl, float* __restrict__ gc) {
  size_t idx = (size_t)blockIdx.x * 256 + threadIdx.x;
  if (idx >= (size_t)(EE + NN) * 64) return;
  int f = (int)(idx & 63);
  int e = (int)(idx >> 6);
  int r, cdst; float nrm;
  if (e < EE) {
    r = ei[e]; cdst = ei[EE + e];
    nrm = dinv[r] * ew[e] * dinv[cdst];
  } else {
    int n = e - EE;
    r = n; cdst = n;
    float di = dinv[n];
    nrm = di * di;
  }
  const float* src = xl + (size_t)r * FOUT + f;
  float* dst = gc + (size_t)cdst * FOUT + f;
#pragma unroll
  for (int g = 0; g < GG; ++g)
    atomAddF(dst + (size_t)g * NN * FOUT, src[(size_t)g * NN * FOUT] * nrm);
}

// ---------------- RNN: persistent over T, 32 rows/block ----------------
__global__ __launch_bounds__(256) void k_rnn(const float* __restrict__ gc,
                                             const float* __restrict__ Wih,
                                             const float* __restrict__ Whh,
                                             const float* __restrict__ bih,
                                             const float* __restrict__ bhh,
                                             float* __restrict__ rnn) {
  extern __shared__ float sm[];
  float* s_wih = sm;                       // 128*64
  float* s_whh = sm + HH * FOUT;           // 128*128
  float* s_h   = s_whh + HH * HH;          // 32*128
  float* s_b   = s_h + 32 * HH;            // 128
  for (int i = threadIdx.x; i < HH * FOUT; i += 256) s_wih[i] = Wih[i];
  for (int i = threadIdx.x; i < HH * HH;  i += 256) s_whh[i] = Whh[i];
  for (int i = threadIdx.x; i < 32 * HH;  i += 256) s_h[i]   = 0.f;
  if (threadIdx.x < HH) s_b[threadIdx.x] = bih[threadIdx.x] + bhh[threadIdx.x];
  __syncthreads();

  const int wave = threadIdx.x >> 5, lane = threadIdx.x & 31;
  const int mr = lane & 15, half = lane >> 4, kk = half * 2;
  const int n0 = wave * 16;                // 8 waves cover 128 cols
  const size_t p0 = (size_t)blockIdx.x * 32;
  const float bias = s_b[n0 + mr];
  const float* bw = s_wih + (n0 + mr) * FOUT + kk;
  const float* bh = s_whh + (n0 + mr) * HH + kk;

  for (int t = 0; t < TT; ++t) {
    v8f c0 = {}, c1 = {};
    // x_t part: seq[p,t,f] = gc_flat[(p*T+t)*64+f]  (raw .view semantics)
    const float* a0 = gc + ((p0 + mr) * TT + t) * FIN + kk;
    const float* a1 = gc + ((p0 + 16 + mr) * TT + t) * FIN + kk;
#pragma unroll
    for (int k = 0; k < FIN; k += 4) {
      v2f b = {bw[k], bw[k + 1]};
      v2f a = {a0[k], a0[k + 1]};
      c0 = wmma4(a, b, c0);
      v2f a2 = {a1[k], a1[k + 1]};
      c1 = wmma4(a2, b, c1);
    }
    // h part from LDS
    const float* h0 = s_h + mr * HH + kk;
    const float* h1 = s_h + (16 + mr) * HH + kk;
#pragma unroll
    for (int k = 0; k < HH; k += 4) {
      v2f b = {bh[k], bh[k + 1]};
      v2f a = {h0[k], h0[k + 1]};
      c0 = wmma4(a, b, c0);
      v2f a2 = {h1[k], h1[k + 1]};
      c1 = wmma4(a2, b, c1);
    }
    __syncthreads();   // everyone done reading s_h
#pragma unroll
    for (int j = 0; j < 8; ++j) {
      int r = j + 8 * half;
      float v0 = tanhf(c0[j] + bias);
      float v1 = tanhf(c1[j] + bias);
      s_h[r * HH + n0 + mr] = v0;
      s_h[(16 + r) * HH + n0 + mr] = v1;
      rnn[((p0 + r) * TT + t) * HH + n0 + mr] = v0;
      rnn[((p0 + 16 + r) * TT + t) * HH + n0 + mr] = v1;
    }
    __syncthreads();   // s_h ready for next step
  }
}

// ---------------- MLP head: 128 -> 512 -> 512 -> 64, 32 rows/block ----------------
__global__ __launch_bounds__(256) void k_mlp(const float* __restrict__ Xg,
                                             const float* __restrict__ W1, const float* __restrict__ B1,
                                             const float* __restrict__ W2, const float* __restrict__ B2,
                                             const float* __restrict__ W3, const float* __restrict__ B3,
                                             float* __restrict__ out) {
  extern __shared__ float sm[];
  float* sX  = sm;                 // 32*128
  float* sY1 = sm + 32 * HH;       // 32*512
  float* sY2 = sY1 + 32 * HID;     // 32*512
  const size_t r0 = (size_t)blockIdx.x * 32;
  for (int i = threadIdx.x; i < 32 * HH; i += 256) sX[i] = Xg[r0 * HH + i];
  __syncthreads();
  const int wave = threadIdx.x >> 5, lane = threadIdx.x & 31;
  const int mr = lane & 15, half = lane >> 4, kk = half * 2;

  // Layer 1: 32x512, 64 tiles, 8 per wave
  for (int ti = 0; ti < 8; ++ti) {
    int tile = wave * 8 + ti;
    int rt = tile & 1, ct = tile >> 1;
    const float* ap = sX + (rt * 16 + mr) * HH + kk;
    const float* bp = W1 + (size_t)(ct * 16 + mr) * HH + kk;
    v8f c = {};
#pragma unroll 8
    for (int k = 0; k < HH; k += 4) {
      v2f a = {ap[k], ap[k + 1]};
      v2f b = {bp[k], bp[k + 1]};
      c = wmma4(a, b, c);
    }
    float bias = B1[ct * 16 + mr];
#pragma unroll
    for (int j = 0; j < 8; ++j) {
      float v = c[j] + bias;
      sY1[(rt * 16 + j + 8 * half) * HID + ct * 16 + mr] = v > 0.f ? v : 0.f;
    }
  }
  __syncthreads();

  // Layer 2: 32x512, K=512
  for (int ti = 0; ti < 8; ++ti) {
    int tile = wave * 8 + ti;
    int rt = tile & 1, ct = tile >> 1;
    const float* ap = sY1 + (rt * 16 + mr) * HID + kk;
    const float* bp = W2 + (size_t)(ct * 16 + mr) * HID + kk;
    v8f c = {};
#pragma unroll 4
    for (int k = 0; k < HID; k += 4) {
      v2f a = {ap[k], ap[k + 1]};
      v2f b = {bp[k], bp[k + 1]};
      c = wmma4(a, b, c);
    }
    float bias = B2[ct * 16 + mr];
#pragma unroll
    for (int j = 0; j < 8; ++j) {
      float v = c[j] + bias;
      sY2[(rt * 16 + j + 8 * half) * HID + ct * 16 + mr] = v > 0.f ? v : 0.f;
    }
  }
  __syncthreads();

  // Layer 3: 32x64, one tile per wave; write to output with view mapping
  {
    int rt = wave & 1, ct = wave >> 1;
    const float* ap = sY2 + (rt * 16 + mr) * HID + kk;
    const float* bp = W3 + (size_t)(ct * 16 + mr) * HID + kk;
    v8f c = {};
#pragma unroll 4
    for (int k = 0; k < HID; k += 4) {
      v2f a = {ap[k], ap[k + 1]};
      v2f b = {bp[k], bp[k + 1]};
      c = wmma4(a, b, c);
    }
    float bias = B3[ct * 16 + mr];
#pragma unroll
    for (int j = 0; j < 8; ++j) {
      size_t r = r0 + rt * 16 + j + 8 * half;   // r = p*T + t
      size_t p = r >> 3;
      int t = (int)(r & 7);
      out[(p * (TT + PP) + t) * FOUT + ct * 16 + mr] = c[j] + bias;
    }
  }
}

// ---------------- zero the P prediction slots ----------------
__global__ void k_zero_pred(float* __restrict__ out) {
  size_t idx = (size_t)blockIdx.x * 256 + threadIdx.x;     // 20000*2*64 items
  if (idx < (size_t)BN * PP * FOUT) {
    size_t p = idx >> 7;          // 128 trailing floats per row-group
    size_t rem = idx & 127;
    out[p * (size_t)(TT + PP) * FOUT + (size_t)TT * FOUT + rem] = 0.f;
  }
}

extern "C" void kernel_launch(void* const* d_in, const int* in_sizes, int n_in,
                              void* d_out, int out_size, void* d_ws, size_t ws_size,
                              hipStream_t stream) {
  const float* x_in  = (const float*)d_in[0];
  const int*   ei    = (const int*)  d_in[1];
  const float* ew    = (const float*)d_in[2];
  const float* gcn_w = (const float*)d_in[3];
  const float* gcn_b = (const float*)d_in[4];
  const float* w_ih  = (const float*)d_in[5];
  const float* w_hh  = (const float*)d_in[6];
  const float* b_ih  = (const float*)d_in[7];
  const float* b_hh  = (const float*)d_in[8];
  const float* w1    = (const float*)d_in[9];
  const float* b1    = (const float*)d_in[10];
  const float* w2    = (const float*)d_in[11];
  const float* b2    = (const float*)d_in[12];
  const float* w3    = (const float*)d_in[13];
  const float* b3    = (const float*)d_in[14];
  float* out = (float*)d_out;

  float* ws   = (float*)d_ws;
  float* deg  = ws;                          // N
  float* dinv = ws + 16384;                  // N
  float* xl   = ws + 32768;                  // G*N*64
  float* gc   = xl + GNF;                    // G*N*64
  float* rnn  = gc + GNF;                    // BN*T*128

  // norm precompute
  k_deg_init<<<(NN + 255) / 256, 256, 0, stream>>>(deg);
  k_deg_scatter<<<(EE + NN + 255) / 256, 256, 0, stream>>>(ei, ew, deg);
  k_dinv<<<(NN + 255) / 256, 256, 0, stream>>>(deg, dinv);

  // GCN linear
  k_gcn_gemm<<<(GG * NN) / 32, 256, 0, stream>>>(x_in, gcn_w, xl);

  // aggregate
  k_agg_init<<<(int)((GNF + 255) / 256), 256, 0, stream>>>(gcn_b, gc);
  k_scatter<<<(int)(((size_t)(EE + NN) * 64 + 255) / 256), 256, 0, stream>>>(
      ei, ew, dinv, xl, gc);

  // RNN (persistent over T)
  size_t rnn_lds = (size_t)(HH * FOUT + HH * HH + 32 * HH + HH) * sizeof(float);
  k_rnn<<<BN / 32, 256, rnn_lds, stream>>>(gc, w_ih, w_hh, b_ih, b_hh, rnn);

  // MLP head
  size_t mlp_lds = (size_t)(32 * HH + 32 * HID + 32 * HID) * sizeof(float);
  k_mlp<<<(BN * TT) / 32, 256, mlp_lds, stream>>>(rnn, w1, b1, w2, b2, w3, b3, out);

  // zero prediction slots
  k_zero_pred<<<(int)(((size_t)BN * PP * FOUT + 255) / 256), 256, 0, stream>>>(out);
}